// AdaPeakConv2D_34316788695841
// MI455X (gfx1250) — compile-verified
//
#include <hip/hip_runtime.h>
#include <hip/hip_bf16.h>
#include <math.h>

#define Bn   8
#define Cc   32
#define Hh   128
#define Ww   128
#define OUTC 32
#define HWp  132              // padded extent (H + 2*PAD), PAD = 2
#define NPRF 16
#define KTOT 512              // C * NPRF
#define PLANE 16384           // H*W
#define NSTAT 131072          // B*H*W elements per BN channel

typedef __attribute__((ext_vector_type(2))) float v2f;
typedef __attribute__((ext_vector_type(8))) float v8f;

// ws layout (floats):
// [0..3]   sum per guide channel (v0,v1,h0,h1)
// [4..7]   sumsq per guide channel
// [8..11]  BN scale per channel
// [12..15] BN shift per channel
// [16..]   raw guide conv outputs: raw[ch][b][i][j], ch-stride = 131072

__global__ void k_zero_sums(float* ws) {
    if (threadIdx.x < 8) ws[threadIdx.x] = 0.0f;
}

// Guide convs (7x1 vertical, 1x7 horizontal, zero-padded) + stat accumulation.
// One block = one (b, i) row, 128 threads (j).
__global__ void k_gb_conv(const float* __restrict__ x,
                          const float* __restrict__ w_vrt, const float* __restrict__ b_vrt,
                          const float* __restrict__ w_hrz, const float* __restrict__ b_hrz,
                          float* __restrict__ ws) {
    __shared__ float swv[2 * Cc * 7];
    __shared__ float swh[2 * Cc * 7];
    __shared__ float acc8[8];
    const int tid = threadIdx.x;
    for (int t = tid; t < 2 * Cc * 7; t += 128) { swv[t] = w_vrt[t]; swh[t] = w_hrz[t]; }
    if (tid < 8) acc8[tid] = 0.0f;
    __syncthreads();

    const int b = blockIdx.x >> 7;
    const int i = blockIdx.x & 127;
    const int j = tid;

    float rv0 = b_vrt[0], rv1 = b_vrt[1];
    float rh0 = b_hrz[0], rh1 = b_hrz[1];
    const float* xb = x + (size_t)b * Cc * PLANE;
    for (int c = 0; c < Cc; ++c) {
        const float* xc = xb + c * PLANE;
        #pragma unroll
        for (int t = 0; t < 7; ++t) {
            int r  = i + t - 3;
            float xv = (r >= 0 && r < Hh) ? xc[r * Ww + j] : 0.0f;
            rv0 += swv[c * 7 + t] * xv;
            rv1 += swv[(Cc + c) * 7 + t] * xv;
            int cl = j + t - 3;
            float xh = (cl >= 0 && cl < Ww) ? xc[i * Ww + cl] : 0.0f;
            rh0 += swh[c * 7 + t] * xh;
            rh1 += swh[(Cc + c) * 7 + t] * xh;
        }
    }

    float* raw = ws + 16;
    const int pix = b * PLANE + i * Ww + j;
    raw[0 * NSTAT + pix] = rv0;
    raw[1 * NSTAT + pix] = rv1;
    raw[2 * NSTAT + pix] = rh0;
    raw[3 * NSTAT + pix] = rh1;

    atomicAdd(&acc8[0], rv0);           atomicAdd(&acc8[1], rv1);
    atomicAdd(&acc8[2], rh0);           atomicAdd(&acc8[3], rh1);
    atomicAdd(&acc8[4], rv0 * rv0);     atomicAdd(&acc8[5], rv1 * rv1);
    atomicAdd(&acc8[6], rh0 * rh0);     atomicAdd(&acc8[7], rh1 * rh1);
    __syncthreads();
    if (tid < 8) atomicAdd(&ws[tid], acc8[tid]);
}

__global__ void k_bn_stats(const float* __restrict__ g_vrt, const float* __restrict__ be_vrt,
                           const float* __restrict__ g_hrz, const float* __restrict__ be_hrz,
                           float* __restrict__ ws) {
    const int ch = threadIdx.x;
    if (ch >= 4) return;
    const float invN = 1.0f / (float)NSTAT;
    float mean = ws[ch] * invN;
    float var  = ws[4 + ch] * invN - mean * mean;
    float gamma = (ch < 2) ? g_vrt[ch]  : g_hrz[ch - 2];
    float beta  = (ch < 2) ? be_vrt[ch] : be_hrz[ch - 2];
    float scale = gamma * rsqrtf(var + 1e-5f);
    ws[8 + ch]  = scale;
    ws[12 + ch] = beta - mean * scale;
}

// Main kernel: deformable ring sampling + per-pixel 32x512 GEMM via WMMA f32 16x16x4.
// One block = one (b, i) row. 256 threads = 8 waves; wave w owns pixels j = 16w..16w+15.
// Lane L: pixel = L%16, half = L/16. Lane precomputes bilinear data for its 8 ring
// taps n = 4*tt + 2*half + {0,1} (exactly the K rows the WMMA B-layout assigns it).
__global__ void k_ada_main(const float* __restrict__ x,
                           const float* __restrict__ w_pk, const float* __restrict__ b_pk,
                           const float* __restrict__ ws,
                           float* __restrict__ out) {
    extern __shared__ float smem[];
    float* wlds = smem;               // 32*512 floats (64 KB): w_pk[o][k]
    float* blds = smem + OUTC * KTOT; // 32 floats: b_pk

    const int tid = threadIdx.x;
    for (int t = tid; t < OUTC * KTOT; t += 256) wlds[t] = w_pk[t];
    if (tid < OUTC) blds[tid] = b_pk[tid];
    __syncthreads();

    const int b    = blockIdx.x >> 7;
    const int i    = blockIdx.x & 127;
    const int wv   = tid >> 5;
    const int lane = tid & 31;
    const int half = lane >> 4;
    const int colp = lane & 15;
    const int j    = wv * 16 + colp;

    // Guide values for this pixel: gb = sigmoid(raw*scale + shift) * 2
    const float* raw = ws + 16;
    const int pix = b * PLANE + i * Ww + j;
    float gb[4];
    #pragma unroll
    for (int ch = 0; ch < 4; ++ch) {
        float t = raw[ch * NSTAT + pix] * ws[8 + ch] + ws[12 + ch];
        gb[ch] = 2.0f / (1.0f + expf(-t));
    }

    const float PRFX[NPRF] = {-2,-2,-2,-2, -2,-1, 0, 1,  2, 2, 2, 2, -1, 0, 1, 2};
    const float PRFY[NPRF] = {-2,-1, 0, 1,  2, 2, 2, 2, -1, 0, 1, 2, -2,-2,-2,-2};

    int   ilt[8], irb[8];
    float glt[8], grb[8];
    #pragma unroll
    for (int s = 0; s < 8; ++s) {
        const int n = 4 * (s >> 1) + 2 * half + (s & 1);
        float mvx = (n < 4) ? -gb[0] : ((n < 8) ? 0.0f : ((n < 12) ? gb[1] : 0.0f));
        float mvy = (n < 4) ? 0.0f   : ((n < 8) ? gb[3] : ((n < 12) ? 0.0f : -gb[2]));
        float px = (float)(i + 2) + PRFX[n] + mvx;
        float py = (float)(j + 2) + PRFY[n] + mvy;
        float qlx = floorf(px), qly = floorf(py);
        float qlxi = fminf(fmaxf(qlx, 0.0f), (float)(HWp - 1));
        float qlyi = fminf(fmaxf(qly, 0.0f), (float)(HWp - 1));
        float qrxi = fminf(fmaxf(qlx + 1.0f, 0.0f), (float)(HWp - 1));
        float qryi = fminf(fmaxf(qly + 1.0f, 0.0f), (float)(HWp - 1));
        float pxc  = fminf(fmaxf(px, 0.0f), (float)(HWp - 1));
        float pyc  = fminf(fmaxf(py, 0.0f), (float)(HWp - 1));
        glt[s] = (1.0f + (qlxi - pxc)) * (1.0f + (qlyi - pyc));
        grb[s] = (1.0f - (qrxi - pxc)) * (1.0f - (qryi - pyc));
        // edge-padded tensor == clamped index into x
        int rlt = min(max((int)qlxi - 2, 0), Hh - 1);
        int clt = min(max((int)qlyi - 2, 0), Ww - 1);
        int rrb = min(max((int)qrxi - 2, 0), Hh - 1);
        int crb = min(max((int)qryi - 2, 0), Ww - 1);
        ilt[s] = rlt * Ww + clt;
        irb[s] = rrb * Ww + crb;
    }

    v8f acc0 = {};  // output channels 0..15
    v8f acc1 = {};  // output channels 16..31
    const int idxc = i * Ww + j;
    const float* xb = x + (size_t)b * Cc * PLANE;

    #pragma unroll 1
    for (int c = 0; c < Cc; ++c) {
        const float* xc = xb + c * PLANE;
        const float xcen = xc[idxc];
        const int kc = c * NPRF;
        #pragma unroll
        for (int tt = 0; tt < 4; ++tt) {
            const int s0 = 2 * tt, s1 = s0 + 1;
            v2f bv;
            bv.x = xcen - (glt[s0] * xc[ilt[s0]] + grb[s0] * xc[irb[s0]]);
            bv.y = xcen - (glt[s1] * xc[ilt[s1]] + grb[s1] * xc[irb[s1]]);
            const int kk = kc + 4 * tt + 2 * half;  // even -> 8B aligned ds_load_b64
            const v2f a0 = *(const v2f*)(wlds + colp * KTOT + kk);
            const v2f a1 = *(const v2f*)(wlds + (colp + 16) * KTOT + kk);
            acc0 = __builtin_amdgcn_wmma_f32_16x16x4_f32(false, a0, false, bv,
                                                         (short)0, acc0, false, false);
            acc1 = __builtin_amdgcn_wmma_f32_16x16x4_f32(false, a1, false, bv,
                                                         (short)0, acc1, false, false);
        }
    }

    // C/D layout: VGPR r holds M = r (lanes 0-15) / M = r+8 (lanes 16-31); N = lane%16.
    float* outb = out + (size_t)b * OUTC * PLANE;
    #pragma unroll
    for (int r = 0; r < 8; ++r) {
        const int o0 = r + 8 * half;
        const int o1 = 16 + o0;
        outb[o0 * PLANE + i * Ww + j] = acc0[r] + blds[o0];
        outb[o1 * PLANE + i * Ww + j] = acc1[r] + blds[o1];
    }
}

extern "C" void kernel_launch(void* const* d_in, const int* in_sizes, int n_in,
                              void* d_out, int out_size, void* d_ws, size_t ws_size,
                              hipStream_t stream) {
    const float* x      = (const float*)d_in[0];
    const float* w_vrt  = (const float*)d_in[1];
    const float* b_vrt  = (const float*)d_in[2];
    const float* g_vrt  = (const float*)d_in[3];
    const float* be_vrt = (const float*)d_in[4];
    const float* w_hrz  = (const float*)d_in[5];
    const float* b_hrz  = (const float*)d_in[6];
    const float* g_hrz  = (const float*)d_in[7];
    const float* be_hrz = (const float*)d_in[8];
    const float* w_pk   = (const float*)d_in[9];
    const float* b_pk   = (const float*)d_in[10];
    float* out = (float*)d_out;
    float* ws  = (float*)d_ws;

    k_zero_sums<<<1, 32, 0, stream>>>(ws);
    k_gb_conv<<<dim3(Bn * Hh), 128, 0, stream>>>(x, w_vrt, b_vrt, w_hrz, b_hrz, ws);
    k_bn_stats<<<1, 32, 0, stream>>>(g_vrt, be_vrt, g_hrz, be_hrz, ws);
    const size_t smem = (size_t)(OUTC * KTOT + OUTC) * sizeof(float); // 64 KB + 128 B
    k_ada_main<<<dim3(Bn * Hh), 256, smem, stream>>>(x, w_pk, b_pk, ws, out);
}